// SelfAttentiveSpanExtractor_26448408609335
// MI455X (gfx1250) — compile-verified
//
#include <hip/hip_runtime.h>

typedef __attribute__((ext_vector_type(2))) float v2f;
typedef __attribute__((ext_vector_type(8))) float v8f;

#define BB 8
#define LL 2048
#define DD 1024
#define NSPAN 512
#define MAXW 32

// ---------------------------------------------------------------------------
// Kernel 1: logits[b*L + l] = seq[b,l,:] . W_att + b_att
// via V_WMMA_F32_16X16X4_F32 (fp32 WMMA -> exact-precision GEMV).
// One wave computes 16 token logits; 8 waves/block -> 128 tokens/block.
// ---------------------------------------------------------------------------
__global__ __launch_bounds__(256) void span_logits_wmma_kernel(
    const float* __restrict__ seq,     // [B*L, D]
    const float* __restrict__ Watt,    // [D]
    const float* __restrict__ batt,    // [1]
    float* __restrict__ logits)        // [B*L]
{
    __shared__ float Wl[DD];
    for (int i = threadIdx.x; i < DD; i += 256) Wl[i] = Watt[i];
    __syncthreads();

    const int wave  = threadIdx.x >> 5;        // 0..7
    const int lane  = threadIdx.x & 31;
    const int m     = lane & 15;               // A-matrix row (token within 16)
    const int khalf = (lane >> 4) << 1;        // K sub-pair: lanes 0-15 -> K 0,1; lanes 16-31 -> K 2,3
    const long t0   = (long)blockIdx.x * 128 + (long)wave * 16;  // first of 16 tokens

    const float* arow = seq + (t0 + m) * (long)DD + khalf;

    v8f c = {};
    #pragma unroll 4
    for (int kb = 0; kb < DD; kb += 4) {
        v2f a, b;
        a.x = arow[kb];
        a.y = arow[kb + 1];
        b.x = Wl[kb + khalf];       // B[k][n] = W[kb+k] broadcast to all 16 columns
        b.y = Wl[kb + khalf + 1];
        // D = A(16x4) * B(4x16) + C, fp32 accumulate
        c = __builtin_amdgcn_wmma_f32_16x16x4_f32(
                /*neg_a=*/false, a, /*neg_b=*/false, b,
                /*c_mod=*/(short)0, c, /*reuse_a=*/false, /*reuse_b=*/false);
    }

    const float bias = batt[0];
    // C/D layout: lane l holds C[M = v + (l>=16 ? 8 : 0)][N = l&15] in c[v].
    // Lanes with N==0 (lanes 0 and 16) jointly hold all 16 logits.
    if ((lane & 15) == 0) {
        const int mbase = (lane >> 4) * 8;
        #pragma unroll
        for (int v = 0; v < 8; ++v)
            logits[t0 + mbase + v] = c[v] + bias;
    }
}

// ---------------------------------------------------------------------------
// Kernel 2: one span per block. Wave 0 reproduces the reference masked
// softmax exactly (softmax over all 32 slots of logit*mask, then *mask,
// then renormalize with +1e-13). Then 256 threads x float4 accumulate the
// weighted sum over the span rows (fully coalesced 4KB row reads, L2 hits).
// ---------------------------------------------------------------------------
__global__ __launch_bounds__(256) void span_weighted_sum_kernel(
    const float* __restrict__ seq,     // [B, L, D]
    const int*   __restrict__ spans,   // [B, N, 2]
    const float* __restrict__ logits,  // [B, L]
    float* __restrict__ out)           // [B, N, D]
{
    __shared__ float att_s[MAXW];

    const int s     = blockIdx.x;          // flattened span id: b*NSPAN + n
    const int b     = s / NSPAN;
    const int start = spans[2 * s];
    const int end   = spans[2 * s + 1];
    const int width = end - start;

    if (threadIdx.x < 32) {
        const int  w   = threadIdx.x;
        const int  raw = end - w;
        const bool in  = (w <= width) && (raw >= 0);
        const int  idx = raw > 0 ? raw : 0;
        const float z  = in ? logits[b * LL + idx] : 0.0f;   // logit*mask

        // softmax over all 32 slots (masked slots contribute exp(0))
        float M = z;
        #pragma unroll
        for (int o = 16; o > 0; o >>= 1) M = fmaxf(M, __shfl_xor(M, o, 32));
        const float e = __expf(z - M);
        float S = e;
        #pragma unroll
        for (int o = 16; o > 0; o >>= 1) S += __shfl_xor(S, o, 32);
        const float p  = e / S;
        const float pm = in ? p : 0.0f;                      // * mask
        float Sm = pm;
        #pragma unroll
        for (int o = 16; o > 0; o >>= 1) Sm += __shfl_xor(Sm, o, 32);
        att_s[w] = pm / (Sm + 1e-13f);                       // renormalize
    }
    __syncthreads();

    const float4* srow = (const float4*)(seq + (long)b * LL * DD);
    const int t = threadIdx.x;                 // d-chunk: covers d = 4t..4t+3
    float4 acc = {0.0f, 0.0f, 0.0f, 0.0f};

    for (int w = 0; w <= width && w < MAXW; ++w) {
        const int raw = end - w;
        const int idx = raw > 0 ? raw : 0;
        const float aw = att_s[w];
        const float4 v = srow[(long)idx * (DD / 4) + t];
        acc.x += aw * v.x;
        acc.y += aw * v.y;
        acc.z += aw * v.z;
        acc.w += aw * v.w;
    }

    ((float4*)out)[(long)s * (DD / 4) + t] = acc;
}

// ---------------------------------------------------------------------------
// Launch: in_sizes order = seq [B*L*D] f32, spans [B*N*2] i32, W_att [D] f32,
// b_att [1] f32. d_out = [B*N*D] f32. d_ws holds logits [B*L] f32 (64 KB).
// ---------------------------------------------------------------------------
extern "C" void kernel_launch(void* const* d_in, const int* in_sizes, int n_in,
                              void* d_out, int out_size, void* d_ws, size_t ws_size,
                              hipStream_t stream) {
    const float* seq   = (const float*)d_in[0];
    const int*   spans = (const int*)d_in[1];
    const float* Watt  = (const float*)d_in[2];
    const float* batt  = (const float*)d_in[3];
    float* out    = (float*)d_out;
    float* logits = (float*)d_ws;

    const int tokens = BB * LL;                 // 16384
    const int blocks1 = tokens / 128;           // 128 blocks (exact fit)
    span_logits_wmma_kernel<<<blocks1, 256, 0, stream>>>(seq, Watt, batt, logits);

    const int blocks2 = BB * NSPAN;             // 4096 spans, one per block
    span_weighted_sum_kernel<<<blocks2, 256, 0, stream>>>(seq, spans, logits, out);
}